// NormalGatedRepulsionLoss_66769561584286
// MI455X (gfx1250) — compile-verified
//
#include <hip/hip_runtime.h>
#include <math.h>

// NormalGatedRepulsionLoss for MI455X (gfx1250).
// Stencil + masked reduction; memory-light (~10MB), transcendental-bound.
// - gfx1250 async global->LDS staging (GLOBAL_LOAD_ASYNC_TO_LDS / ASYNCcnt)
// - normalize each voxel once in LDS, 124-tap stencil out of LDS
// - exact wave32 reductions on the matrix pipe via V_WMMA_F32_16X16X4_F32
//   (A = lane data in documented 16x4 f32 layout, B = all-ones -> row sums),
//   deterministic 2-stage global sum (no float atomics).

#define TAU   2
#define TILE  8
#define HD    12                 // TILE + 2*TAU
#define HVOX  (HD * HD * HD)     // 1728 halo voxels
#define NOFF  124
#define NBLK  1024               // 2 batches * (64/8)^3 tiles

typedef __attribute__((ext_vector_type(2))) float v2f;
typedef __attribute__((ext_vector_type(8))) float v8f;

// Exact 32-lane sum using the matrix pipe:
// A[m][k] layout (ISA 7.12.2, 32-bit A 16x4): lane m: VGPR0=K0, VGPR1=K1;
// lane m+16: VGPR0=K2, VGPR1=K3.  With A VGPR0 = s, VGPR1 = 0 and B == 1
// everywhere (layout-independent), D[m][n] = s_m + s_{m+16} for every n.
// D layout: VGPR r holds row r (lanes 0-15) / row r+8 (lanes 16-31), so
// summing the 8 accumulators + one xor-16 shuffle yields the full wave sum.
__device__ __forceinline__ float wave_reduce_sum_wmma(float s)
{
    v2f a; a.x = s;    a.y = 0.0f;
    v2f b; b.x = 1.0f; b.y = 1.0f;
    v8f c = {};
    c = __builtin_amdgcn_wmma_f32_16x16x4_f32(false, a, false, b,
                                              (short)0, c, false, false);
    float r = ((c[0] + c[1]) + (c[2] + c[3])) + ((c[4] + c[5]) + (c[6] + c[7]));
    r += __shfl_xor(r, 16, 32);
    return r;
}

__global__ __launch_bounds__(256)
void ngr_main_kernel(const float* __restrict__ n_pred,
                     const float* __restrict__ src_pred,
                     float* __restrict__ ws_tot,
                     unsigned* __restrict__ ws_cnt)
{
    __shared__ float    stage[4 * HVOX];   // raw nx | ny | nz | src planes
    __shared__ float4   tile[HVOX];        // normalized n + mask
    __shared__ int      s_doff[NOFF];      // halo-linear offset per stencil tap
    __shared__ float    s_wd[NOFF];        // spatial gaussian weight
    __shared__ float    s_wtot[8];
    __shared__ unsigned s_wcnt[8];

    const int tid = threadIdx.x;
    const unsigned bidx = blockIdx.x;
    const int b  = (int)(bidx >> 9);
    const int z0 = (int)((bidx >> 6) & 7u) * TILE - TAU;
    const int y0 = (int)((bidx >> 3) & 7u) * TILE - TAU;
    const int x0 = (int)( bidx        & 7u) * TILE - TAU;

    // ---- stencil tap table (124 taps; float64 exp -> f32 like numpy) ----
    if (tid < NOFF) {
        int t  = (tid < 62) ? tid : tid + 1;          // skip center (t==62)
        int dz = t / 25 - 2;
        int dy = (t / 5) % 5 - 2;
        int dx = t % 5 - 2;
        s_doff[tid] = (dz * HD + dy) * HD + dx;
        double r2 = (double)(dz * dz + dy * dy + dx * dx);
        s_wd[tid] = (float)exp(-r2 * (9.0 / 16.0));   // SIGMA_D2 = (2/1.5)^2
    }

    // ---- async global -> LDS staging of the 12^3 halo tile ----
    const unsigned cs   = 64u * 64u * 64u * 4u;       // channel stride, bytes
    const unsigned lds0 = (unsigned)(size_t)(&stage[0]);
    for (int i = tid; i < HVOX; i += 256) {
        int hz = i / (HD * HD);
        int hy = (i / HD) % HD;
        int hx = i % HD;
        int gz = z0 + hz, gy = y0 + hy, gx = x0 + hx;
        // clamp for a safe (dummy) load; out-of-bounds zeroed in pass 2
        int cz = min(max(gz, 0), 63);
        int cy = min(max(gy, 0), 63);
        int cx = min(max(gx, 0), 63);
        unsigned vox  = (((unsigned)cz * 64u + (unsigned)cy) * 64u + (unsigned)cx) * 4u;
        unsigned offn = (unsigned)b * (3u * cs) + vox;   // n_pred[b,0,z,y,x]
        unsigned offs = (unsigned)b * cs + vox;          // src  [b,0,z,y,x]
        unsigned l0   = lds0 + (unsigned)i * 4u;
        asm volatile("global_load_async_to_lds_b32 %0, %1, %2"
                     :: "v"(l0),                  "v"(offn),           "s"(n_pred) : "memory");
        asm volatile("global_load_async_to_lds_b32 %0, %1, %2"
                     :: "v"(l0 + 1u * HVOX * 4u), "v"(offn + cs),      "s"(n_pred) : "memory");
        asm volatile("global_load_async_to_lds_b32 %0, %1, %2"
                     :: "v"(l0 + 2u * HVOX * 4u), "v"(offn + 2u * cs), "s"(n_pred) : "memory");
        asm volatile("global_load_async_to_lds_b32 %0, %1, %2"
                     :: "v"(l0 + 3u * HVOX * 4u), "v"(offs),           "s"(src_pred) : "memory");
    }
    asm volatile("s_wait_asynccnt 0" ::: "memory");
    __syncthreads();

    // ---- normalize once per halo voxel; zero-pad out-of-bounds ----
    for (int i = tid; i < HVOX; i += 256) {
        int hz = i / (HD * HD);
        int hy = (i / HD) % HD;
        int hx = i % HD;
        int gz = z0 + hz, gy = y0 + hy, gx = x0 + hx;
        bool inb = ((unsigned)gz < 64u) & ((unsigned)gy < 64u) & ((unsigned)gx < 64u);
        float4 o = make_float4(0.f, 0.f, 0.f, 0.f);
        if (inb) {
            float nx = stage[i];
            float ny = stage[HVOX + i];
            float nz = stage[2 * HVOX + i];
            float sv = stage[3 * HVOX + i];
            float nr = sqrtf(nx * nx + ny * ny + nz * nz);
            float sc = 1.0f / fmaxf(nr, 1e-6f);
            o = make_float4(nx * sc, ny * sc, nz * sc,
                            (sv > 0.0f) ? 1.0f : 0.0f);  // sigmoid(x)>0.5 <=> x>0
        }
        tile[i] = o;
    }
    __syncthreads();

    // ---- 124-tap stencil over 8^3 interior (2 voxels / thread) ----
    const float inv_sth2 = (float)(1.0 / (0.3490658503988659 * 0.3490658503988659));
    float    tot = 0.0f;
    unsigned cnt = 0u;
    for (int v = tid; v < TILE * TILE * TILE; v += 256) {
        int lz = v >> 6, ly = (v >> 3) & 7, lx = v & 7;
        int hc = ((lz + TAU) * HD + (ly + TAU)) * HD + (lx + TAU);
        float4 c = tile[hc];
        if (c.w > 0.5f) {
            #pragma unroll 4
            for (int o = 0; o < NOFF; ++o) {
                float4 nb = tile[hc + s_doff[o]];
                float  d  = fminf(fmaxf(c.x * nb.x + c.y * nb.y + c.z * nb.z, -1.0f), 1.0f);
                float  th = acosf(d);
                float  wt = expf(-th * th * inv_sth2);
                tot += s_wd[o] * wt * nb.w;          // nb.w in {0,1}
                cnt += (nb.w > 0.5f) ? 1u : 0u;
            }
        }
    }

    // ---- exact wave32 reduction on the matrix pipe (EXEC all-ones here) ----
    // per-wave cnt <= 8*1024 << 2^24, so f32 WMMA accumulation is bit-exact.
    float wtot = wave_reduce_sum_wmma(tot);
    float wcnt = wave_reduce_sum_wmma((float)cnt);

    int wave = tid >> 5, lane = tid & 31;
    if (lane == 0) {
        s_wtot[wave] = wtot;
        s_wcnt[wave] = (unsigned)(wcnt + 0.5f);
    }
    __syncthreads();
    if (tid == 0) {
        float T = 0.f; unsigned C = 0u;
        #pragma unroll
        for (int w = 0; w < 8; ++w) { T += s_wtot[w]; C += s_wcnt[w]; }
        ws_tot[bidx] = T;
        ws_cnt[bidx] = C;
    }
}

__global__ __launch_bounds__(256)
void ngr_final_kernel(const float* __restrict__ ws_tot,
                      const unsigned* __restrict__ ws_cnt,
                      float* __restrict__ out)
{
    __shared__ float    st[8];
    __shared__ unsigned sc[8];
    float    t = 0.f;
    unsigned c = 0u;
    for (int i = threadIdx.x; i < NBLK; i += 256) {
        t += ws_tot[i];
        c += ws_cnt[i];
    }
    // totals via matrix-pipe reduction; counts via exact integer shuffles
    t = wave_reduce_sum_wmma(t);
    for (int off = 16; off > 0; off >>= 1) c += __shfl_down(c, off, 32);

    if ((threadIdx.x & 31) == 0) { st[threadIdx.x >> 5] = t; sc[threadIdx.x >> 5] = c; }
    __syncthreads();
    if (threadIdx.x == 0) {
        float T = 0.f; unsigned C = 0u;
        #pragma unroll
        for (int w = 0; w < 8; ++w) { T += st[w]; C += sc[w]; }
        out[0] = (C != 0u) ? (T / (float)C) : 0.0f;
    }
}

extern "C" void kernel_launch(void* const* d_in, const int* in_sizes, int n_in,
                              void* d_out, int out_size, void* d_ws, size_t ws_size,
                              hipStream_t stream)
{
    const float* n_pred   = (const float*)d_in[0];   // (2,3,64,64,64) f32
    const float* src_pred = (const float*)d_in[1];   // (2,1,64,64,64) f32
    float*    ws_tot = (float*)d_ws;                           // NBLK floats
    unsigned* ws_cnt = (unsigned*)((char*)d_ws + NBLK * sizeof(float));

    ngr_main_kernel<<<NBLK, 256, 0, stream>>>(n_pred, src_pred, ws_tot, ws_cnt);
    ngr_final_kernel<<<1, 256, 0, stream>>>(ws_tot, ws_cnt, (float*)d_out);
}